// IrisQNNClassifier_730144440593
// MI455X (gfx1250) — compile-verified
//
#include <hip/hip_runtime.h>
#include <math.h>

typedef float v2f __attribute__((ext_vector_type(2)));
typedef float v8f __attribute__((ext_vector_type(8)));

#define QNN_L 8
#define QNN_NQ 4

// ws layout (floats):
//   [0 .. 1023]   four 16x16 QNN matrices Q_k, row-major, k-major
//   [1024..1027]  w_k = ent0[k]^2 for k = 0..3
//   [1028]        w_id = sum_{k=4..7} ent0[k]^2

__global__ __launch_bounds__(256) void build_qnn_kernel(const float* __restrict__ alpha,
                                                        const float* __restrict__ thetas,
                                                        float* __restrict__ ws) {
    __shared__ float U[16][17];
    __shared__ float T[16][17];
    const int t = threadIdx.x;
    const int i = t >> 4, j = t & 15;
    const int blk = blockIdx.x;
    const float* th = thetas + blk * (QNN_L * QNN_NQ);

    U[i][j] = (i == j) ? 1.f : 0.f;
    __syncthreads();

    // Entangler permutation f(i): CNOT(0,1), CNOT(1,2), CNOT(2,3), qubit0 = MSB.
    int b0 = (i >> 3) & 1, b1 = (i >> 2) & 1, b2 = (i >> 1) & 1, b3 = i & 1;
    if (b0) b1 ^= 1;
    if (b1) b2 ^= 1;
    if (b2) b3 ^= 1;
    const int fi = (b0 << 3) | (b1 << 2) | (b2 << 1) | b3;

    for (int l = 0; l < QNN_L; ++l) {
        float c[4], s[4];
        for (int q = 0; q < 4; ++q) {
            float a = th[l * QNN_NQ + q] * 0.5f;
            sincosf(a, &s[q], &c[q]);
        }
        // tmp[i][j] = sum_kk R[i][kk] * U[kk][j], R = kron of 4 RY matrices
        float acc = 0.f;
        for (int kk = 0; kk < 16; ++kk) {
            float r = 1.f;
            for (int q = 0; q < 4; ++q) {
                int bi = (i >> (3 - q)) & 1;
                int bk = (kk >> (3 - q)) & 1;
                // RY = [[c, -s], [s, c]]
                float f = bi ? (bk ? c[q] : s[q]) : (bk ? -s[q] : c[q]);
                r *= f;
            }
            acc += r * U[kk][j];
        }
        __syncthreads();
        T[fi][j] = acc;          // U <- ENT * (R * U): row f(i) gets row i
        __syncthreads();
        U[i][j] = T[i][j];
        __syncthreads();
    }
    ws[blk * 256 + i * 16 + j] = U[i][j];

    if (blk == 0 && t == 0) {
        // ent0 = layered ancilla product applied to e0 (8-vector)
        float v[8] = {1.f, 0.f, 0.f, 0.f, 0.f, 0.f, 0.f, 0.f};
        int idx = 0;
        for (int l = 0; l < 3; ++l) {
            int m = 1 << (2 - l);
            float nv[8];
            for (int ii = 0; ii < 8; ++ii) {
                int r = ii >> (2 - l);        // row within block-diag
                int p = ii & (m - 1);         // position within trailing identity
                int b = r >> 1, rb = r & 1;   // RY block, row-in-block
                float cs, sn;
                sincosf(alpha[idx + b] * 0.5f, &sn, &cs);
                float f0 = rb ? sn : cs;      // RY[rb][0]
                float f1 = rb ? cs : -sn;     // RY[rb][1]
                nv[ii] = f0 * v[(2 * b) * m + p] + f1 * v[(2 * b + 1) * m + p];
            }
            for (int ii = 0; ii < 8; ++ii) v[ii] = nv[ii];
            idx += (1 << l);
        }
        for (int k = 0; k < 4; ++k) ws[1024 + k] = v[k] * v[k];
        ws[1028] = v[4] * v[4] + v[5] * v[5] + v[6] * v[6] + v[7] * v[7];
    }
}

// lane-xor-16 exchange: ds_swizzle group-of-32, xor_mask=0x10, and_mask=0x1f
__device__ __forceinline__ float swap16(float v) {
    return __int_as_float(__builtin_amdgcn_ds_swizzle(__float_as_int(v), 0x401F));
}

__global__ __launch_bounds__(256) void qnn_batch_kernel(const float* __restrict__ x,
                                                        const float* __restrict__ ws,
                                                        float* __restrict__ out) {
    const int lane = threadIdx.x & 31;
    const int g = (blockIdx.x * blockDim.x + threadIdx.x) >> 5;  // global wave id
    const int m0 = g * 16;                                       // batch tile base
    const int n = lane & 15;                                     // M for A (Q rows), N for B/C (batch col)
    const int half = lane >> 4;

    // ---- encode state amplitudes for this lane's batch row ----
    const float4 xv = reinterpret_cast<const float4*>(x)[m0 + n];
    float c0, s0, c1, s1, c2, s2, c3, s3;
    {
        const float TWO_PI = 6.28318530717958647692f;
        sincosf((xv.x * TWO_PI) * 0.5f, &s0, &c0);
        sincosf((xv.y * TWO_PI) * 0.5f, &s1, &c1);
        sincosf((xv.z * TWO_PI) * 0.5f, &s2, &c2);
        sincosf((xv.w * TWO_PI) * 0.5f, &s3, &c3);
    }
    float p01[4] = {c0 * c1, c0 * s1, s0 * c1, s0 * s1};
    float p23[4] = {c2 * c3, c2 * s3, s2 * c3, s2 * s3};

    // B fragments = ENC^T: lane holds B[j = ks*4 + 2*half + {0,1}][b = n] = enc[n][j]
    v2f ef[4];
    const float q0 = p23[2 * half], q1 = p23[2 * half + 1];
    for (int ks = 0; ks < 4; ++ks) {
        ef[ks].x = p01[ks] * q0;
        ef[ks].y = p01[ks] * q1;
    }

    // A fragments = Q_k: lane holds A[i = n][ks*4 + 2*half + {0,1}]
    v2f qa[4][4];
    for (int k = 0; k < 4; ++k)
        for (int ks = 0; ks < 4; ++ks) {
            const float* p = ws + k * 256 + n * 16 + ks * 4 + 2 * half;
            qa[k][ks].x = p[0];
            qa[k][ks].y = p[1];
        }
    const float w0 = ws[1024], w1 = ws[1025], w2 = ws[1026], w3 = ws[1027];
    const float wid = ws[1028];

    // ---- D_k = Q_k * ENC^T : four 16x16x16 products as 16x16x4 f32 WMMAs ----
    // C layout: lane holds D[i = r + 8*half][b = n] in acc[r]
    v8f acc0 = {}, acc1 = {}, acc2 = {}, acc3 = {};
    for (int ks = 0; ks < 4; ++ks)
        acc0 = __builtin_amdgcn_wmma_f32_16x16x4_f32(false, qa[0][ks], false, ef[ks],
                                                     (short)0, acc0, false, false);
    for (int ks = 0; ks < 4; ++ks)
        acc1 = __builtin_amdgcn_wmma_f32_16x16x4_f32(false, qa[1][ks], false, ef[ks],
                                                     (short)0, acc1, false, false);
    for (int ks = 0; ks < 4; ++ks)
        acc2 = __builtin_amdgcn_wmma_f32_16x16x4_f32(false, qa[2][ks], false, ef[ks],
                                                     (short)0, acc2, false, false);
    for (int ks = 0; ks < 4; ++ks)
        acc3 = __builtin_amdgcn_wmma_f32_16x16x4_f32(false, qa[3][ks], false, ef[ks],
                                                     (short)0, acc3, false, false);

    // ---- weighted squares: S[r] = sum_k w_k * D_k[i=r+8*half][b]^2 ----
    float S[8];
    for (int r = 0; r < 8; ++r)
        S[r] = w0 * acc0[r] * acc0[r] + w1 * acc1[r] * acc1[r] +
               w2 * acc2[r] * acc2[r] + w3 * acc3[r] * acc3[r];

    // Z-signed sums over state index i = r + 8*half (all in-register):
    //   e3: sign = bit3(i) = half      e4: sign = bit2(i) = r>>2      e5: sign = bit1(i) = r>>1
    const float s01 = S[0] + S[1], s23 = S[2] + S[3];
    const float s45 = S[4] + S[5], s67 = S[6] + S[7];
    const float lo = s01 + s23, hi = s45 + s67;
    float P3 = lo + hi;
    float P4 = lo - hi;
    float P5 = (s01 - s23) + (s45 - s67);
    P3 = half ? -P3 : P3;

    // combine the two halves (i in 0..7 with i in 8..15): one xor-16 exchange each
    const float e3 = P3 + swap16(P3);
    const float e4 = P4 + swap16(P4);
    const float e5 = P5 + swap16(P5);

    if (half == 0) {
        // identity-block contribution: w_id * cos(2*pi*x_q) = w_id*(c^2 - s^2)
        const float id3 = wid * (c0 * c0 - s0 * s0);
        const float id4 = wid * (c1 * c1 - s1 * s1);
        const float id5 = wid * (c2 * c2 - s2 * s2);
        float* o = out + (size_t)(m0 + n) * 3;
        o[0] = (e3 + id3 + 1.f) * 0.5f;
        o[1] = (e4 + id4 + 1.f) * 0.5f;
        o[2] = (e5 + id5 + 1.f) * 0.5f;
    }
}

extern "C" void kernel_launch(void* const* d_in, const int* in_sizes, int n_in,
                              void* d_out, int out_size, void* d_ws, size_t ws_size,
                              hipStream_t stream) {
    const float* x      = (const float*)d_in[0];
    const float* alpha  = (const float*)d_in[1];
    const float* thetas = (const float*)d_in[2];
    float* ws  = (float*)d_ws;
    float* o   = (float*)d_out;

    const int batch = in_sizes[0] / 4;        // rows of x
    build_qnn_kernel<<<4, 256, 0, stream>>>(alpha, thetas, ws);
    // 8 waves per block, 16 batch rows per wave -> 128 rows per block
    const int blocks = batch / 128;
    qnn_batch_kernel<<<blocks, 256, 0, stream>>>(x, ws, o);
}